// PyHashGridBG_57853209477254
// MI455X (gfx1250) — compile-verified
//
#include <hip/hip_runtime.h>
#include <hip/hip_bf16.h>

// Instant-NGP hash-grid encode, tuned for MI455X (gfx1250, wave32).
//
// Mapping: one thread per (point n, level l); g = n*16 + l.
//  - 16 lanes of a wave half share a point -> cooperative xyz fetch +
//    ds_bpermute broadcast (wave32-aware, offloads the vmem pipe).
//  - 8 independent global_load_b64 gathers per thread (MLP=8) from the
//    64 MB table, which is L2-resident (192 MB L2); a prefetch kernel
//    (global_prefetch_b8) warms it.
//  - Output (128 MB, write-once) uses nontemporal stores so the streaming
//    writes don't evict the table from L2.

typedef __attribute__((ext_vector_type(2))) float f2;

#define NGP_LEVELS 16u
#define NGP_LOG2T 19u
#define NGP_TMASK ((1u << NGP_LOG2T) - 1u)
#define NGP_P1 2654435761u
#define NGP_P2 805459861u

__global__ __launch_bounds__(256) void ngp_prefetch_table(const char* __restrict__ feat,
                                                          unsigned nbytes) {
  unsigned i = blockIdx.x * 256u + threadIdx.x;
  unsigned off = i * 128u;  // one prefetch per 128B line
  if (off < nbytes) {
    __builtin_prefetch(feat + off, 0, 3);  // read, keep in far cache (L2)
  }
}

__global__ __launch_bounds__(256) void ngp_hashgrid(const float* __restrict__ x,
                                                    const float* __restrict__ features,
                                                    const int* __restrict__ resolution,
                                                    float* __restrict__ out,
                                                    unsigned N) {
  const unsigned g = blockIdx.x * 256u + threadIdx.x;
  const unsigned l = g & 15u;       // level
  const unsigned n = g >> 4;        // point index

  // ---- cooperative coordinate fetch (all 32 lanes participate) ----
  const unsigned lane = threadIdx.x & 31u;
  const unsigned sub  = lane & 15u;             // == l
  const unsigned half = lane & 16u;             // which point in the wave
  float t = 0.0f;
  if (sub < 3u && n < N) {
    // coords are read exactly once kernel-wide: nontemporal load
    t = __builtin_nontemporal_load(x + (n * 3u + sub));
  }
  const float px = __shfl(t, (int)(half + 0u), 32);  // -> ds_bpermute_b32
  const float py = __shfl(t, (int)(half + 1u), 32);
  const float pz = __shfl(t, (int)(half + 2u), 32);

  if (n >= N) return;

  // ---- level-scaled coords, base corner, trilinear weights ----
  const float res = (float)resolution[l];
  const float xs = px * res, ys = py * res, zs = pz * res;
  const float fx = floorf(xs), fy = floorf(ys), fz = floorf(zs);
  const float wx = xs - fx, wy = ys - fy, wz = zs - fz;
  const unsigned x0 = (unsigned)fx, y0 = (unsigned)fy, z0 = (unsigned)fz;

  // ---- hash components (strength-reduced: p*(k+1) = p*k + p) ----
  const unsigned hx0 = x0,            hx1 = x0 + 1u;            // prime 1
  const unsigned hy0 = y0 * NGP_P1,   hy1 = hy0 + NGP_P1;
  const unsigned hz0 = z0 * NGP_P2,   hz1 = hz0 + NGP_P2;

  const f2* __restrict__ tbl = (const f2*)features + ((size_t)l << NGP_LOG2T);

  // corner c: bit0->x offset, bit1->y, bit2->z (reference order c=0..7)
  const unsigned h0 = (hx0 ^ hy0 ^ hz0) & NGP_TMASK;  // (0,0,0)
  const unsigned h1 = (hx1 ^ hy0 ^ hz0) & NGP_TMASK;  // (1,0,0)
  const unsigned h2 = (hx0 ^ hy1 ^ hz0) & NGP_TMASK;  // (0,1,0)
  const unsigned h3 = (hx1 ^ hy1 ^ hz0) & NGP_TMASK;  // (1,1,0)
  const unsigned h4 = (hx0 ^ hy0 ^ hz1) & NGP_TMASK;  // (0,0,1)
  const unsigned h5 = (hx1 ^ hy0 ^ hz1) & NGP_TMASK;  // (1,0,1)
  const unsigned h6 = (hx0 ^ hy1 ^ hz1) & NGP_TMASK;  // (0,1,1)
  const unsigned h7 = (hx1 ^ hy1 ^ hz1) & NGP_TMASK;  // (1,1,1)

  // issue all 8 gathers before any use: MLP = 8 (global_load_b64 each)
  const f2 c0 = tbl[h0];
  const f2 c1 = tbl[h1];
  const f2 c2 = tbl[h2];
  const f2 c3 = tbl[h3];
  const f2 c4 = tbl[h4];
  const f2 c5 = tbl[h5];
  const f2 c6 = tbl[h6];
  const f2 c7 = tbl[h7];

  // ---- trilinear weights (shared y*z products) ----
  const float wx0 = 1.0f - wx;
  const float wy0 = 1.0f - wy;
  const float wz0 = 1.0f - wz;
  const float a00 = wy0 * wz0;  // y=0,z=0
  const float a10 = wy  * wz0;  // y=1,z=0
  const float a01 = wy0 * wz;   // y=0,z=1
  const float a11 = wy  * wz;   // y=1,z=1

  const float w0 = wx0 * a00, w1 = wx * a00;
  const float w2 = wx0 * a10, w3 = wx * a10;
  const float w4 = wx0 * a01, w5 = wx * a01;
  const float w6 = wx0 * a11, w7 = wx * a11;

  // accumulate in reference corner order
  float o0 = c0.x * w0;
  float o1 = c0.y * w0;
  o0 = fmaf(c1.x, w1, o0);  o1 = fmaf(c1.y, w1, o1);
  o0 = fmaf(c2.x, w2, o0);  o1 = fmaf(c2.y, w2, o1);
  o0 = fmaf(c3.x, w3, o0);  o1 = fmaf(c3.y, w3, o1);
  o0 = fmaf(c4.x, w4, o0);  o1 = fmaf(c4.y, w4, o1);
  o0 = fmaf(c5.x, w5, o0);  o1 = fmaf(c5.y, w5, o1);
  o0 = fmaf(c6.x, w6, o0);  o1 = fmaf(c6.y, w6, o1);
  o0 = fmaf(c7.x, w7, o0);  o1 = fmaf(c7.y, w7, o1);

  // out[n, l*2 + f] == ((f2*)out)[n*16 + l] == ((f2*)out)[g]
  f2 o;
  o.x = o0;
  o.y = o1;
  __builtin_nontemporal_store(o, (f2*)out + g);  // don't pollute L2
}

extern "C" void kernel_launch(void* const* d_in, const int* in_sizes, int n_in,
                              void* d_out, int out_size, void* d_ws, size_t ws_size,
                              hipStream_t stream) {
  const float* x          = (const float*)d_in[0];   // [N,3] f32
  const float* features   = (const float*)d_in[1];   // [16, 2^19, 2] f32
  const int*   resolution = (const int*)d_in[2];     // [16] i32
  float*       out        = (float*)d_out;           // [N, 32] f32

  const unsigned N = (unsigned)(in_sizes[0] / 3);

  // Warm L2 with the 64 MB feature table (one global_prefetch_b8 / 128B line).
  const unsigned fbytes = (unsigned)in_sizes[1] * 4u;
  const unsigned lines  = (fbytes + 127u) / 128u;
  ngp_prefetch_table<<<(lines + 255u) / 256u, 256, 0, stream>>>(
      (const char*)features, fbytes);

  const unsigned total = N * NGP_LEVELS;  // one thread per (point, level)
  ngp_hashgrid<<<(total + 255u) / 256u, 256, 0, stream>>>(
      x, features, resolution, out, N);
}